// BCosGraphSAGE_28346784153652
// MI455X (gfx1250) — compile-verified
//
#include <hip/hip_runtime.h>
#include <hip/hip_bf16.h>

typedef float v2f __attribute__((ext_vector_type(2)));
typedef float v8f __attribute__((ext_vector_type(8)));

#define EPS      1e-6f
#define NORM_EPS 1e-12f

__device__ __forceinline__ void atomAddF32(float* p, float v) {
    __hip_atomic_fetch_add(p, v, __ATOMIC_RELAXED, __HIP_MEMORY_SCOPE_AGENT);
}

// ---------------------------------------------------------------- utilities
__global__ void zero_f32(float* __restrict__ p, int n) {
    int t = blockIdx.x * blockDim.x + threadIdx.x;
    if (t < n) p[t] = 0.0f;
}

// deg[row[e]] += 1
__global__ void deg_count(const long long* __restrict__ rows, float* __restrict__ deg, int E) {
    int e = blockIdx.x * blockDim.x + threadIdx.x;
    if (e < E) atomAddF32(&deg[(int)rows[e]], 1.0f);
}

// deg <- 1 / max(deg, 1)
__global__ void deg_finalize(float* __restrict__ deg, int N) {
    int n = blockIdx.x * blockDim.x + threadIdx.x;
    if (n < N) deg[n] = 1.0f / fmaxf(deg[n], 1.0f);
}

// r[n] = 1 / max(||X[n,:]||, 1e-12)   (rows of length 64)
__global__ void rownorm_inv(const float* __restrict__ X, float* __restrict__ r, int N) {
    int n = blockIdx.x * blockDim.x + threadIdx.x;
    if (n >= N) return;
    const float4* p = (const float4*)(X + (size_t)n * 64);
    float s = 0.0f;
#pragma unroll
    for (int i = 0; i < 16; ++i) {
        float4 v = p[i];
        s += v.x * v.x + v.y * v.y + v.z * v.z + v.w * v.w;
    }
    r[n] = 1.0f / fmaxf(sqrtf(s), NORM_EPS);
}

// ---------------------------------------------------------------- WMMA GEMM
// One block = 16 nodes. 8 waves: waves 0-3 -> bcos(X @ Wn^T) col-tiles 0..3,
// waves 4-7 -> X @ Ws^T col-tiles 0..3. K = 64 via 16x V_WMMA_F32_16X16X4_F32.
// A layout (16x4 f32): lanes 0-15 hold M=lane with K = k0+{0,1};
//                      lanes 16-31 hold M=lane-16 with K = k0+{2,3}.
// B layout mirrors A with N per lane; B[k][n] = W[n][k] (row-major W rows).
__global__ void gemm_bcos_self(const float* __restrict__ X,   // [N,64]
                               const float* __restrict__ Wn,  // [64,64] neigh
                               const float* __restrict__ Ws,  // [64,64] self
                               const float* __restrict__ rx,  // [N]
                               const float* __restrict__ rw,  // [64] (neigh rows)
                               float* __restrict__ Y,         // [N,64] bcos out
                               float* __restrict__ S,         // [N,64] self out
                               int N) {
    const int wave = threadIdx.x >> 5;
    const int lane = threadIdx.x & 31;
    const int m0   = blockIdx.x * 16;
    const bool selfpath = (wave >= 4);
    const int n0   = (wave & 3) * 16;

    const int half = lane >> 4;          // 0 or 1
    const int l16  = lane & 15;
    const int koff = half * 2;           // K sub-offset within a 4-block

    int mrow = m0 + l16; if (mrow >= N) mrow = N - 1;   // keep EXEC all-ones
    const float* arow = X + (size_t)mrow * 64;
    const float* brow = (selfpath ? Ws : Wn) + (size_t)(n0 + l16) * 64;

    v8f c = {};
#pragma unroll
    for (int k0 = 0; k0 < 64; k0 += 4) {
        v2f a = *(const v2f*)(arow + k0 + koff);
        v2f b = *(const v2f*)(brow + k0 + koff);
        c = __builtin_amdgcn_wmma_f32_16x16x4_f32(
                /*neg_a=*/false, a, /*neg_b=*/false, b,
                /*c_mod=*/(short)0, c, /*reuse_a=*/false, /*reuse_b=*/false);
    }

    // C/D layout: VGPR r, lanes 0-15 -> (M=r, N=l16); lanes 16-31 -> (M=8+r, N=l16)
    const int ncol  = n0 + l16;
    const int mbase = m0 + half * 8;
    const float rwn = selfpath ? 1.0f : rw[ncol];
#pragma unroll
    for (int r = 0; r < 8; ++r) {
        const int m = mbase + r;
        if (m >= N) continue;
        const float lin = c[r];
        if (selfpath) {
            S[(size_t)m * 64 + ncol] = lin;
        } else {
            float cosv = fmaxf(lin * rx[m] * rwn, EPS);
            Y[(size_t)m * 64 + ncol] = lin * cosv;
        }
    }
}

// ---------------------------------------------------------------- scatter
// 16 threads per edge; each gathers a float4 of Y[col] and atomically adds
// into agg[row]. Within a wave: 2 edges, fully coalesced 256B gathers.
__global__ void scatter_add(const long long* __restrict__ rows,
                            const long long* __restrict__ cols,
                            const float* __restrict__ Y,
                            float* __restrict__ agg, int E) {
    int t = blockIdx.x * blockDim.x + threadIdx.x;
    int e = t >> 4;
    if (e >= E) return;
    int seg = (t & 15) << 2;  // 0,4,...,60
    long long r = rows[e];
    long long cidx = cols[e];
    const float4 v = *(const float4*)(Y + (size_t)cidx * 64 + seg);
    float* dst = agg + (size_t)r * 64 + seg;
    atomAddF32(dst + 0, v.x);
    atomAddF32(dst + 1, v.y);
    atomAddF32(dst + 2, v.z);
    atomAddF32(dst + 3, v.w);
}

// h = relu(agg * deg_inv + s);  agg <- 0 (ready for layer 2)
__global__ void relu_combine(float* __restrict__ h, float* __restrict__ agg,
                             const float* __restrict__ dinv, int total) {
    int t = blockIdx.x * blockDim.x + threadIdx.x;
    if (t >= total) return;
    int n = t >> 6;
    float v = agg[t] * dinv[n] + h[t];
    h[t] = fmaxf(v, 0.0f);
    agg[t] = 0.0f;
}

// out = agg * deg_inv + out   (out already holds the self-path)
__global__ void out_combine(float* __restrict__ out, const float* __restrict__ agg,
                            const float* __restrict__ dinv, int total) {
    int t = blockIdx.x * blockDim.x + threadIdx.x;
    if (t >= total) return;
    int n = t >> 6;
    out[t] = agg[t] * dinv[n] + out[t];
}

// ---------------------------------------------------------------- launcher
extern "C" void kernel_launch(void* const* d_in, const int* in_sizes, int n_in,
                              void* d_out, int out_size, void* d_ws, size_t ws_size,
                              hipStream_t stream) {
    const float*     x   = (const float*)d_in[0];
    const long long* ei  = (const long long*)d_in[1];   // [2, E] int64
    const float*     wn1 = (const float*)d_in[2];
    const float*     wn2 = (const float*)d_in[3];
    const float*     ws1 = (const float*)d_in[4];
    const float*     ws2 = (const float*)d_in[5];
    float*           out = (float*)d_out;

    const int N = in_sizes[0] / 64;
    const int E = in_sizes[1] / 2;
    const long long* rows = ei;
    const long long* cols = ei + E;

    float* ws   = (float*)d_ws;
    float* dinv = ws;                       // N
    float* rx   = ws + N;                   // N
    float* rw1  = ws + 2 * (size_t)N;       // 64
    float* rw2  = rw1 + 64;                 // 64
    float* y    = rw2 + 64;                 // N*64
    float* h    = y + (size_t)N * 64;       // N*64
    float* agg  = h + (size_t)N * 64;       // N*64

    const int NT = 256;
    const int NV = N * 64;
    dim3 blkNV((NV + NT - 1) / NT), blkN((N + NT - 1) / NT);
    dim3 blkE((E + NT - 1) / NT), blkE16(((size_t)E * 16 + NT - 1) / NT);
    dim3 blkG(N / 16 + (N % 16 != 0));

    // degree + inverse norms
    zero_f32<<<blkN, NT, 0, stream>>>(dinv, N);
    zero_f32<<<blkNV, NT, 0, stream>>>(agg, NV);
    deg_count<<<blkE, NT, 0, stream>>>(rows, dinv, E);
    deg_finalize<<<blkN, NT, 0, stream>>>(dinv, N);
    rownorm_inv<<<1, 64, 0, stream>>>(wn1, rw1, 64);
    rownorm_inv<<<1, 64, 0, stream>>>(wn2, rw2, 64);

    // layer 1
    rownorm_inv<<<blkN, NT, 0, stream>>>(x, rx, N);
    gemm_bcos_self<<<blkG, 256, 0, stream>>>(x, wn1, ws1, rx, rw1, y, h, N);
    scatter_add<<<blkE16, NT, 0, stream>>>(rows, cols, y, agg, E);
    relu_combine<<<blkNV, NT, 0, stream>>>(h, agg, dinv, NV);

    // layer 2
    rownorm_inv<<<blkN, NT, 0, stream>>>(h, rx, N);
    gemm_bcos_self<<<blkG, 256, 0, stream>>>(h, wn2, ws2, rx, rw2, y, out, N);
    scatter_add<<<blkE16, NT, 0, stream>>>(rows, cols, y, agg, E);
    out_combine<<<blkNV, NT, 0, stream>>>(out, agg, dinv, NV);
}